// HierarchicalPooling_89404039234014
// MI455X (gfx1250) — compile-verified
//
#include <hip/hip_runtime.h>
#include <hip/hip_bf16.h>
#include <math.h>

// Problem constants (from reference)
#define NN 32768
#define DD 256
#define HH 8
#define DHH 32
#define SS 4
#define CSZ 3276          // cluster size = N // 10
#define CC 11             // num clusters = ceil(N / CSZ)
#define SHCOLS 32         // S*H score columns
#define SPLIT 32          // stats stage-1 slices per cluster
#define KSPLIT 13         // k_xw K-dimension split (52 chunks / 13 = 4 per block)

typedef __attribute__((ext_vector_type(2))) float v2f;
typedef __attribute__((ext_vector_type(8))) float v8f;

static __device__ __forceinline__ v8f wmma4(v2f a, v2f b, v8f c) {
    // V_WMMA_F32_16X16X4_F32 : D(16x16 f32) = A(16x4 f32) * B(4x16 f32) + C
    return __builtin_amdgcn_wmma_f32_16x16x4_f32(false, a, false, b, (short)0, c,
                                                 false, false);
}

// ---------------------------------------------------------------------------
// Kernel 1: fold pool_query into Wk:  wq[sh][d] = scale * <Wk[s,d,h*32:], q[s,h,:]>
//           sbias[sh] = scale * <bk[s,h*32:], q[s,h,:]>
// grid = 32 blocks (one per sh), 256 threads (one per d)
// ---------------------------------------------------------------------------
__global__ void k_prep(const float* __restrict__ Wk, const float* __restrict__ bk,
                       const float* __restrict__ q,
                       float* __restrict__ wq, float* __restrict__ sbias) {
    const int sh = blockIdx.x;          // 0..31
    const int s  = sh >> 3;
    const int h  = sh & 7;
    const int d  = threadIdx.x;         // 0..255
    const float scale = 0.17677669529663687f;   // 32^-0.5

    const float* wrow = Wk + s * (DD * DD) + d * DD + h * DHH;
    const float* qrow = q + s * DD + h * DHH;
    float acc = 0.f;
    #pragma unroll
    for (int dh = 0; dh < DHH; ++dh) acc += wrow[dh] * qrow[dh];
    wq[sh * DD + d] = acc * scale;

    if (d == 0) {
        const float* brow = bk + s * DD + h * DHH;
        float b = 0.f;
        #pragma unroll
        for (int dh = 0; dh < DHH; ++dh) b += brow[dh] * qrow[dh];
        sbias[sh] = b * scale;
    }
}

// ---------------------------------------------------------------------------
// Kernel 2: scores[n, sh] = x[n,:] @ wq[sh,:] + sbias[sh]
// WMMA fp32: M = 16 node rows per wave, N = 32 sh cols (2 tiles), K = 256 (step 4)
// grid = N/128 blocks, 256 threads (8 waves x 16 rows = 128 rows / block)
// ---------------------------------------------------------------------------
__global__ void k_scores(const float* __restrict__ x, const float* __restrict__ wq,
                         const float* __restrict__ sbias, float* __restrict__ scores) {
    const int tid  = threadIdx.x;
    const int lane = tid & 31;
    const int wv   = tid >> 5;
    const int l16  = lane & 15;
    const int half = lane >> 4;
    const int koff = half * 2;

    const int rbase = blockIdx.x * 128 + wv * 16;
    const float* xrow  = x  + (size_t)(rbase + l16) * DD;  // A: M = l16
    const float* wqr0  = wq + (size_t)(l16) * DD;          // B tile0: sh = l16
    const float* wqr1  = wq + (size_t)(16 + l16) * DD;     // B tile1: sh = 16+l16

    v8f acc0, acc1;
    const float sb0 = sbias[l16];
    const float sb1 = sbias[16 + l16];
    #pragma unroll
    for (int v = 0; v < 8; ++v) { acc0[v] = sb0; acc1[v] = sb1; }

    #pragma unroll 4
    for (int k0 = 0; k0 < DD; k0 += 4) {
        v2f a  = *(const v2f*)(xrow + k0 + koff);
        v2f b0 = *(const v2f*)(wqr0 + k0 + koff);
        v2f b1 = *(const v2f*)(wqr1 + k0 + koff);
        acc0 = wmma4(a, b0, acc0);
        acc1 = wmma4(a, b1, acc1);
    }

    #pragma unroll
    for (int v = 0; v < 8; ++v) {
        const int m = v + half * 8;
        scores[(size_t)(rbase + m) * SHCOLS + l16]      = acc0[v];
        scores[(size_t)(rbase + m) * SHCOLS + 16 + l16] = acc1[v];
    }
}

// ---------------------------------------------------------------------------
// Kernel 3a: per-slice softmax stats (stage 1).
// grid = (C, SPLIT) blocks, 256 threads: col = tid%32, 8-way row parallelism.
// Each block covers ~1/SPLIT of its cluster; emits slice max + slice sum(exp).
// ---------------------------------------------------------------------------
__global__ void k_stats1(const float* __restrict__ scores,
                         float* __restrict__ pmax, float* __restrict__ psum) {
    __shared__ float red[256];
    const int c   = blockIdx.x;
    const int p   = blockIdx.y;
    const int tid = threadIdx.x;
    const int col = tid & 31;
    const int rp  = tid >> 5;        // 0..7
    const int start = c * CSZ;
    const int end   = min(NN, start + CSZ);
    const int cnt   = end - start;
    const int s0 = start + (cnt * p) / SPLIT;
    const int s1 = start + (cnt * (p + 1)) / SPLIT;

    float m = -3.402823e38f;
    for (int n = s0 + rp; n < s1; n += 8)
        m = fmaxf(m, scores[(size_t)n * SHCOLS + col]);
    red[tid] = m;
    __syncthreads();
    if (rp == 0) {
        #pragma unroll
        for (int j = 1; j < 8; ++j) m = fmaxf(m, red[j * 32 + col]);
        red[col] = m;
    }
    __syncthreads();
    m = red[col];

    float ssum = 0.f;
    for (int n = s0 + rp; n < s1; n += 8)
        ssum += expf(scores[(size_t)n * SHCOLS + col] - m);
    __syncthreads();                 // all reads of red[col] done before reuse
    red[tid] = ssum;
    __syncthreads();
    if (rp == 0) {
        #pragma unroll
        for (int j = 1; j < 8; ++j) ssum += red[j * 32 + col];
        pmax[((size_t)c * SPLIT + p) * 32 + col] = m;
        psum[((size_t)c * SPLIT + p) * 32 + col] = ssum;
    }
}

// ---------------------------------------------------------------------------
// Kernel 3b: combine slice stats -> cmax[c,col], cinv[c,col] = 1/sum(exp)
// grid = C blocks, 32 threads (one wave), log-sum-exp merge.
// ---------------------------------------------------------------------------
__global__ void k_stats2(const float* __restrict__ pmax, const float* __restrict__ psum,
                         float* __restrict__ cmax, float* __restrict__ cinv) {
    const int c   = blockIdx.x;
    const int col = threadIdx.x;     // 0..31
    float m = -3.402823e38f;
    #pragma unroll 4
    for (int p = 0; p < SPLIT; ++p)
        m = fmaxf(m, pmax[((size_t)c * SPLIT + p) * 32 + col]);
    float ssum = 0.f;
    #pragma unroll 4
    for (int p = 0; p < SPLIT; ++p) {
        const float pm = pmax[((size_t)c * SPLIT + p) * 32 + col];
        ssum += psum[((size_t)c * SPLIT + p) * 32 + col] * expf(pm - m);
    }
    cmax[c * 32 + col] = m;
    cinv[c * 32 + col] = 1.0f / ssum;
}

// ---------------------------------------------------------------------------
// Kernel 4: partial xw: for K-slice p of cluster c:
//   xwpart[c, p, sh, e] = sum_{n in slice} softmax_w[n,sh] * x[n,e]
// Per block: M=32 (sh, 2 tiles), N=256 (8 waves x 32 e-cols), K = 4 chunks of 64.
// Weights staged per 64-node chunk into LDS (stride 68 -> conflict-free A reads).
// grid = (C, KSPLIT) blocks, 256 threads (8 waves)
// ---------------------------------------------------------------------------
__global__ void k_xw(const float* __restrict__ x, const float* __restrict__ scores,
                     const float* __restrict__ cmax, const float* __restrict__ cinv,
                     float* __restrict__ xwpart) {
    __shared__ float wlds[32 * 68];
    const int c    = blockIdx.x;
    const int p    = blockIdx.y;
    const int tid  = threadIdx.x;
    const int lane = tid & 31;
    const int wv   = tid >> 5;
    const int l16  = lane & 15;
    const int half = lane >> 4;

    const int start = c * CSZ;
    const int end   = min(NN, start + CSZ);
    const int nChunks = (end - start + 63) >> 6;
    const int chb = (nChunks + KSPLIT - 1) / KSPLIT;      // chunks per block
    const int ch0 = p * chb;
    const int ch1 = min(nChunks, ch0 + chb);

    const int   ssh = tid & 31;                 // staging column
    const float cm  = cmax[c * 32 + ssh];
    const float ci  = cinv[c * 32 + ssh];

    const int e0 = wv * 32;                     // this wave's 32 e-columns
    v8f acc00, acc01, acc10, acc11;
    #pragma unroll
    for (int v = 0; v < 8; ++v) { acc00[v] = 0.f; acc01[v] = 0.f; acc10[v] = 0.f; acc11[v] = 0.f; }

    for (int ch = ch0; ch < ch1; ++ch) {
        const int nb = start + ch * 64;
        // stage softmax weights for 64 nodes x 32 cols (w=0 for pad rows)
        #pragma unroll
        for (int i = 0; i < 8; ++i) {
            const int flat = tid + i * 256;
            const int kk   = flat >> 5;         // 0..63
            const int node = nb + kk;
            float w = 0.f;
            if (node < end) {
                w = expf(scores[(size_t)node * SHCOLS + ssh] - cm) * ci;
                __builtin_prefetch(&scores[(size_t)(node + 64) * SHCOLS + ssh], 0, 0);
            }
            wlds[ssh * 68 + kk] = w;
        }
        __syncthreads();

        #pragma unroll 4
        for (int k0 = 0; k0 < 64; k0 += 4) {
            v2f a0 = *(const v2f*)(&wlds[(l16)      * 68 + k0 + half * 2]);
            v2f a1 = *(const v2f*)(&wlds[(16 + l16) * 68 + k0 + half * 2]);
            int n0 = nb + k0 + half * 2;
            int n1 = n0 + 1;
            n0 = min(n0, NN - 1);               // safe clamp; w=0 kills pad rows
            n1 = min(n1, NN - 1);
            const float* r0 = x + (size_t)n0 * DD;
            const float* r1 = x + (size_t)n1 * DD;
            v2f b0, b1;
            b0[0] = r0[e0 + l16];       b0[1] = r1[e0 + l16];
            b1[0] = r0[e0 + 16 + l16];  b1[1] = r1[e0 + 16 + l16];
            acc00 = wmma4(a0, b0, acc00);
            acc01 = wmma4(a0, b1, acc01);
            acc10 = wmma4(a1, b0, acc10);
            acc11 = wmma4(a1, b1, acc11);
        }
        __syncthreads();
    }

    // store partials (blocks past the cluster's chunk range store zeros)
    float* dst = xwpart + ((size_t)c * KSPLIT + p) * 32 * DD;
    #pragma unroll
    for (int v = 0; v < 8; ++v) {
        const int m = v + half * 8;
        dst[(size_t)(m)      * DD + e0 + l16]      = acc00[v];
        dst[(size_t)(m)      * DD + e0 + 16 + l16] = acc01[v];
        dst[(size_t)(16 + m) * DD + e0 + l16]      = acc10[v];
        dst[(size_t)(16 + m) * DD + e0 + 16 + l16] = acc11[v];
    }
}

// ---------------------------------------------------------------------------
// Kernel 5: per-cluster tail. Reduces the KSPLIT xw partials while staging to
// LDS (fixed-order sum -> deterministic), then pooled -> out_proj -> fusion MLP.
// grid = C blocks, 256 threads
// ---------------------------------------------------------------------------
__global__ void k_fuse(const float* __restrict__ xwpart,
                       const float* __restrict__ Wv, const float* __restrict__ bv,
                       const float* __restrict__ Wo, const float* __restrict__ bo,
                       const float* __restrict__ Wf1, const float* __restrict__ bf1,
                       const float* __restrict__ ln_g, const float* __restrict__ ln_b,
                       const float* __restrict__ Wf2, const float* __restrict__ bf2,
                       float* __restrict__ fusedC) {
    __shared__ float xws[32 * 260];
    __shared__ float pooled[SS * DD];
    __shared__ float comb[SS * DD];
    __shared__ float hs[DD];
    __shared__ float red[DD];

    const int c   = blockIdx.x;
    const int tid = threadIdx.x;
    const int start = c * CSZ;
    const int end   = min(NN, start + CSZ);
    const float invCnt = 1.0f / (float)(end - start);

    // reduce K-split partials while staging xw into LDS
    for (int i = tid; i < 32 * DD; i += 256) {
        const int sh = i >> 8, e = i & 255;
        float a = 0.f;
        #pragma unroll
        for (int p = 0; p < KSPLIT; ++p)
            a += xwpart[(((size_t)c * KSPLIT + p) * 32 + sh) * DD + e];
        xws[sh * 260 + e] = a;
    }
    __syncthreads();

    // pooled[s,d] = (xw[sh(d)] . Wv[s,:,d] + bv[s,d]) / count
    for (int s = 0; s < SS; ++s) {
        const int d  = tid;
        const int sh = s * HH + (d >> 5);
        const float* wvcol = Wv + (size_t)s * DD * DD + d;   // stride DD
        const float* xr = &xws[sh * 260];
        float acc = 0.f;
        for (int e = 0; e < DD; ++e) acc += xr[e] * wvcol[(size_t)e * DD];
        pooled[s * DD + d] = (acc + bv[s * DD + d]) * invCnt;
    }
    __syncthreads();

    // combined[s,e] = pooled[s,:] @ Wo[s,:,e] + bo[s,e]
    for (int s = 0; s < SS; ++s) {
        const float* wocol = Wo + (size_t)s * DD * DD + tid;
        const float* pr = &pooled[s * DD];
        float acc = bo[s * DD + tid];
        for (int d = 0; d < DD; ++d) acc += pr[d] * wocol[(size_t)d * DD];
        comb[s * DD + tid] = acc;
    }
    __syncthreads();

    // h = combined @ Wf1 + bf1
    float hval = bf1[tid];
    for (int k = 0; k < SS * DD; ++k) hval += comb[k] * Wf1[(size_t)k * DD + tid];

    // LayerNorm over 256
    red[tid] = hval; __syncthreads();
    for (int off = 128; off > 0; off >>= 1) {
        if (tid < off) red[tid] += red[tid + off];
        __syncthreads();
    }
    const float mu = red[0] * (1.0f / 256.0f);
    __syncthreads();
    const float dv = hval - mu;
    red[tid] = dv * dv; __syncthreads();
    for (int off = 128; off > 0; off >>= 1) {
        if (tid < off) red[tid] += red[tid + off];
        __syncthreads();
    }
    const float var = red[0] * (1.0f / 256.0f);
    const float hn = dv * rsqrtf(var + 1e-5f) * ln_g[tid] + ln_b[tid];
    // exact GELU
    const float g = 0.5f * hn * (1.0f + erff(hn * 0.70710678118654752f));
    hs[tid] = g;
    __syncthreads();

    float acc = bf2[tid];
    for (int j = 0; j < DD; ++j) acc += hs[j] * Wf2[(size_t)j * DD + tid];
    fusedC[c * DD + tid] = acc;
}

// ---------------------------------------------------------------------------
// Kernel 6: broadcast the 11 fused rows to all N output rows (float4)
// ---------------------------------------------------------------------------
__global__ void k_bcast(const float* __restrict__ fusedC, float* __restrict__ out) {
    const int idx = (blockIdx.x * 256 + threadIdx.x) * 4;  // element index
    const int n = idx >> 8;
    const int d = idx & 255;
    const int c = n / CSZ;
    *(float4*)(out + idx) = *(const float4*)(fusedC + c * DD + d);
}

// ---------------------------------------------------------------------------
extern "C" void kernel_launch(void* const* d_in, const int* in_sizes, int n_in,
                              void* d_out, int out_size, void* d_ws, size_t ws_size,
                              hipStream_t stream) {
    const float* x   = (const float*)d_in[0];
    // d_in[1] = edge_index (unused), d_in[2] = batch (unused)
    const float* Wk  = (const float*)d_in[3];
    const float* bk  = (const float*)d_in[4];
    const float* Wv  = (const float*)d_in[5];
    const float* bv  = (const float*)d_in[6];
    const float* Wo  = (const float*)d_in[7];
    const float* bo  = (const float*)d_in[8];
    const float* pq  = (const float*)d_in[9];
    const float* Wf1 = (const float*)d_in[10];
    const float* bf1 = (const float*)d_in[11];
    const float* lng = (const float*)d_in[12];
    const float* lnb = (const float*)d_in[13];
    const float* Wf2 = (const float*)d_in[14];
    const float* bf2 = (const float*)d_in[15];
    float* out = (float*)d_out;

    float* ws     = (float*)d_ws;
    float* wq     = ws;                              // 32*256 = 8192
    float* sbias  = ws + 8192;                       // 32 (pad to 128)
    float* scores = ws + 8320;                       // N*32 = 1048576
    float* cmax   = scores + (size_t)NN * 32;        // C*32 (pad 384)
    float* cinv   = cmax + 384;                      // C*32 (pad 384)
    float* pmax   = cinv + 384;                      // C*SPLIT*32 = 11264
    float* psum   = pmax + (size_t)CC * SPLIT * 32;  // 11264
    float* xwpart = psum + (size_t)CC * SPLIT * 32;  // C*KSPLIT*32*256 = 1171456
    float* fusedC = xwpart + (size_t)CC * KSPLIT * 32 * DD;  // C*256 = 2816
    // total ~9.0 MB of f32 workspace

    k_prep  <<<SHCOLS,   256, 0, stream>>>(Wk, bk, pq, wq, sbias);
    k_scores<<<NN / 128, 256, 0, stream>>>(x, wq, sbias, scores);
    k_stats1<<<dim3(CC, SPLIT),  256, 0, stream>>>(scores, pmax, psum);
    k_stats2<<<CC,        32, 0, stream>>>(pmax, psum, cmax, cinv);
    k_xw    <<<dim3(CC, KSPLIT), 256, 0, stream>>>(x, scores, cmax, cinv, xwpart);
    k_fuse  <<<CC,       256, 0, stream>>>(xwpart, Wv, bv, Wo, bo,
                                           Wf1, bf1, lng, lnb, Wf2, bf2, fusedC);
    k_bcast <<<(NN * DD) / 1024, 256, 0, stream>>>(fusedC, out);
}